// CombinedLoss_49667001811727
// MI455X (gfx1250) — compile-verified
//
#include <hip/hip_runtime.h>

#define BATCH 16
#define NCLS 3
#define HH 512
#define WW 512
#define HW (HH * WW)
#define SENT 0x3FFFFFFF

#define TILE 32
#define HALO 34
#define LDS_PITCH 36          // padded row pitch (ints) to spread banks
#define LOCAL_ITERS 6         // LDS propagation steps per launch
#define GLOBAL_ITERS 12       // global launches (each advances ~6 + 2 jumps)

#if defined(__HIP_DEVICE_COMPILE__) && defined(__gfx1250__) && \
    __has_builtin(__builtin_amdgcn_global_load_async_to_lds_b32)
#define HAVE_ASYNC 1
#else
#define HAVE_ASYNC 0
#endif

typedef __attribute__((address_space(1))) int as_global_int;
typedef __attribute__((address_space(3))) int as_shared_int;

typedef __attribute__((ext_vector_type(2))) float v2f;
typedef __attribute__((ext_vector_type(8))) float v8f;

// Full-wave (32-lane) f32 sum on the FP32 WMMA pipe.
// WMMA1: A[m] = (x[m], 0, x[m+16], 0), B = ones -> D[m][n] = x[m]+x[m+16].
// Fold D's 8 VGPRs (rows m and m+8) into a new A, WMMA2 -> full 32-lane total.
// Call sites are fully convergent (EXEC all-ones required by WMMA).
__device__ __forceinline__ float wave_reduce(float x) {
#if defined(__gfx1250__) && __has_builtin(__builtin_amdgcn_wmma_f32_16x16x4_f32)
  v2f a;    a.x = x;    a.y = 0.0f;
  v2f ones; ones.x = 1.0f; ones.y = 1.0f;
  v8f c = {};
  v8f d1 = __builtin_amdgcn_wmma_f32_16x16x4_f32(false, a, false, ones,
                                                 (short)0, c, false, false);
  v2f a2;
  a2.x = d1[0] + d1[2] + d1[4] + d1[6];
  a2.y = d1[1] + d1[3] + d1[5] + d1[7];
  v8f d2 = __builtin_amdgcn_wmma_f32_16x16x4_f32(false, a2, false, ones,
                                                 (short)0, c, false, false);
  return d2[0];
#else
  for (int off = 16; off >= 1; off >>= 1) x += __shfl_xor(x, off, 32);
  return x;
#endif
}

// ---- accumulator slot map (floats at ws+0) ----
// [0] sum(w*logp_t)   [1] sum(w)   [2] focal numerator  [3] valid count
// [4] pred_bin count  [8+b*3+c] inter  [56+b*3+c] sum probs  [104+b*3+c] one-hot count
// [152+b] separation penalty per batch

__global__ void k_zero(float* acc) {
  for (int j = threadIdx.x; j < 1024; j += 256) acc[j] = 0.0f;
}

__global__ void k_init_pairs(int* vmx_t, int* vmn_t, int* vmx_p, int* vmn_p) {
  size_t i = (size_t)blockIdx.y * HW + (size_t)blockIdx.x * blockDim.x + threadIdx.x;
  vmx_t[i] = 0; vmn_t[i] = SENT;
  vmx_p[i] = 0; vmn_p[i] = SENT;
}

// Fused: softmax (C=3), weighted CE terms, focal, dice partials, binary-mask
// seeds for CC. One block row per batch image so per-(b,c) partials stay in
// registers; wave-reduce 14 accumulators via WMMA, lane 0 does global atomics.
__global__ void k_main(const float* __restrict__ pred, const int* __restrict__ tgt,
                       const float* __restrict__ cw, float* __restrict__ acc,
                       int* __restrict__ t_lab, int* __restrict__ p_lab) {
  const int b = blockIdx.y;
  const int tid = blockIdx.x * blockDim.x + threadIdx.x;
  const int stride = gridDim.x * blockDim.x;               // divides HW exactly
  const size_t ibase = (size_t)b * HW;
  const size_t pbase = (size_t)b * NCLS * HW;
  const float w0 = cw[0], w1 = cw[1], w2 = cw[2];

  float s_wlp = 0.f, s_w = 0.f, foc = 0.f, vld = 0.f, prc = 0.f;
  float it0 = 0.f, it1 = 0.f, it2 = 0.f;
  float up0 = 0.f, up1 = 0.f, up2 = 0.f;
  float oh0 = 0.f, oh1 = 0.f, oh2 = 0.f;

  for (int p = tid; p < HW; p += stride) {
    float x0 = pred[pbase + p];
    float x1 = pred[pbase + HW + p];
    float x2 = pred[pbase + 2 * HW + p];
    float m = fmaxf(x0, fmaxf(x1, x2));
    float e0 = __expf(x0 - m), e1 = __expf(x1 - m), e2 = __expf(x2 - m);
    float se = e0 + e1 + e2;
    float inv = 1.0f / se;
    float p0 = e0 * inv, p1 = e1 * inv, p2 = e2 * inv;
    float lse = m + __logf(se);

    int tv = tgt[ibase + p];
    int tc = min(max(tv, 0), NCLS - 1);
    float validf = (tv != 255) ? 1.f : 0.f;
    float cwt = (tc == 0) ? w0 : ((tc == 1) ? w1 : w2);
    float lpt = ((tc == 0) ? x0 : ((tc == 1) ? x1 : x2)) - lse;
    float wt = cwt * validf;
    float pt = __expf(lpt);
    float om = 1.f - pt;

    s_wlp += wt * lpt;
    s_w   += wt;
    foc   += -wt * om * om * lpt;   // gamma = 2
    vld   += validf;

    up0 += p0; up1 += p1; up2 += p2;
    float is0 = (tv == 0) ? 1.f : 0.f;
    float is1 = (tv == 1) ? 1.f : 0.f;
    float is2 = (tv == 2) ? 1.f : 0.f;
    oh0 += is0; oh1 += is1; oh2 += is2;
    it0 += p0 * is0; it1 += p1 * is1; it2 += p2 * is2;

    int pb = (p2 > 0.5f) ? 1 : 0;
    prc += (float)pb;
    t_lab[ibase + p] = (tv == 2) ? (p + 1) : 0;   // raw pixel-seed labels
    p_lab[ibase + p] = pb ? (p + 1) : 0;
  }

  float vals[14] = {s_wlp, s_w, foc, vld, prc,
                    it0, it1, it2, up0, up1, up2, oh0, oh1, oh2};
#pragma unroll
  for (int k = 0; k < 14; ++k) vals[k] = wave_reduce(vals[k]);

  if ((threadIdx.x & 31) == 0) {
    atomicAdd(&acc[0], vals[0]);
    atomicAdd(&acc[1], vals[1]);
    atomicAdd(&acc[2], vals[2]);
    atomicAdd(&acc[3], vals[3]);
    atomicAdd(&acc[4], vals[4]);
#pragma unroll
    for (int c2 = 0; c2 < 3; ++c2) {
      atomicAdd(&acc[8   + b * 3 + c2], vals[5 + c2]);
      atomicAdd(&acc[56  + b * 3 + c2], vals[8 + c2]);
      atomicAdd(&acc[104 + b * 3 + c2], vals[11 + c2]);
    }
  }
}

// LDS-tiled CC relaxation: stage 34x34 halo tile (async-to-LDS when the
// toolchain exposes it), run LOCAL_ITERS 8-neighbor max iterations entirely in
// LDS, then two long-range pointer-jumps against global labels and write back.
// Monotone non-decreasing everywhere (stale halos / cross-block races benign);
// fixed point = per-component max seed, identical partition to the reference.
__global__ void k_cc(int* __restrict__ labs) {
  __shared__ int tile[HALO * LDS_PITCH];
  int* lab = labs + (size_t)blockIdx.y * HW;
  const int tx = (blockIdx.x & 15) * TILE;
  const int ty = (blockIdx.x >> 4) * TILE;
  const int t = threadIdx.x;

  // ---- stage halo tile (clamp-to-edge: duplicates are harmless under max) ----
  for (int s = t; s < HALO * HALO; s += 256) {
    int row = s / HALO, col = s - row * HALO;
    int gy = min(max(ty + row - 1, 0), HH - 1);
    int gx = min(max(tx + col - 1, 0), WW - 1);
#if HAVE_ASYNC
    __builtin_amdgcn_global_load_async_to_lds_b32(
        (as_global_int*)&lab[gy * WW + gx],
        (as_shared_int*)&tile[row * LDS_PITCH + col],
        0, 0);
#else
    tile[row * LDS_PITCH + col] = lab[gy * WW + gx];
#endif
  }
#if HAVE_ASYNC
#if __has_builtin(__builtin_amdgcn_s_wait_asynccnt)
  __builtin_amdgcn_s_wait_asynccnt(0);
#else
  asm volatile("s_wait_asynccnt 0x0" ::: "memory");
#endif
#endif
  __syncthreads();

  // ---- local propagation in LDS: 4 pixels per thread (32x32 interior) ----
  const int lx = t & 31;
  const int ly0 = t >> 5;            // 0..7; rows ly0, ly0+8, ly0+16, ly0+24
  for (int it = 0; it < LOCAL_ITERS; ++it) {
    int nv[4];
#pragma unroll
    for (int k = 0; k < 4; ++k) {
      const int y = ly0 + 8 * k;
      const int idx = (y + 1) * LDS_PITCH + (lx + 1);
      int c = tile[idx];
      int m2 = c;
      if (c != 0) {
        m2 = max(m2, tile[idx - LDS_PITCH - 1]);
        m2 = max(m2, tile[idx - LDS_PITCH]);
        m2 = max(m2, tile[idx - LDS_PITCH + 1]);
        m2 = max(m2, tile[idx - 1]);
        m2 = max(m2, tile[idx + 1]);
        m2 = max(m2, tile[idx + LDS_PITCH - 1]);
        m2 = max(m2, tile[idx + LDS_PITCH]);
        m2 = max(m2, tile[idx + LDS_PITCH + 1]);
      }
      nv[k] = m2;
    }
    __syncthreads();
#pragma unroll
    for (int k = 0; k < 4; ++k) {
      const int y = ly0 + 8 * k;
      const int idx = (y + 1) * LDS_PITCH + (lx + 1);
      if (nv[k] != 0) tile[idx] = nv[k];
    }
    __syncthreads();
  }

  // ---- long-range pointer-jumps (seed pixels are always in-mask) + writeback
  int cur[4];
#pragma unroll
  for (int k = 0; k < 4; ++k) {
    const int y = ly0 + 8 * k;
    cur[k] = tile[(y + 1) * LDS_PITCH + (lx + 1)];
    if (cur[k] > 0) __builtin_prefetch(&lab[cur[k] - 1], 0, 3);
  }
#pragma unroll
  for (int k = 0; k < 4; ++k) {
    int c = cur[k];
    if (c > 0) {
      c = max(c, lab[c - 1]);
      c = max(c, lab[c - 1]);
    }
    lab[(ty + ly0 + 8 * k) * WW + tx + lx] = c;
  }
}

// Segment min/max of val-labels keyed by key-label via atomics (index label-1).
__global__ void k_scatter(const int* __restrict__ t_lab, const int* __restrict__ p_lab,
                          int* vmx_t, int* vmn_t, int* vmx_p, int* vmn_p) {
  const size_t bb = (size_t)blockIdx.y * HW;
  const size_t i = bb + (size_t)blockIdx.x * blockDim.x + threadIdx.x;
  const int t = t_lab[i], p = p_lab[i];
  if (t > 0 && p > 0) {
    atomicMax(&vmx_t[bb + t - 1], p);
    atomicMin(&vmn_t[bb + t - 1], p);
    atomicMax(&vmx_p[bb + p - 1], t);
    atomicMin(&vmn_p[bb + p - 1], t);
  }
}

// Count labels overlapped by >1 distinct opposing label (min written && max!=min).
__global__ void k_count(const int* __restrict__ vmx_t, const int* __restrict__ vmn_t,
                        const int* __restrict__ vmx_p, const int* __restrict__ vmn_p,
                        float* __restrict__ acc) {
  const int b = blockIdx.y;
  const int* vmx = blockIdx.z ? vmx_p : vmx_t;
  const int* vmn = blockIdx.z ? vmn_p : vmn_t;
  const size_t i = (size_t)b * HW + (size_t)blockIdx.x * blockDim.x + threadIdx.x;
  const int mn = vmn[i], mx = vmx[i];
  float hit = (mn != SENT && mx != mn) ? 1.f : 0.f;
  float s = wave_reduce(hit);     // fully convergent up to here
  if ((threadIdx.x & 31) == 0 && s != 0.f) atomicAdd(&acc[152 + b], s);
}

__global__ void k_final(const float* __restrict__ acc, float* __restrict__ out) {
  if (threadIdx.x != 0) return;
  float ce = -acc[0] / acc[1];
  float dsum = 0.f;
  for (int b = 0; b < BATCH; ++b)
    for (int c = 0; c < NCLS; ++c) {
      float inter = acc[8 + b * 3 + c];
      float upv   = acc[56 + b * 3 + c];
      float ohv   = acc[104 + b * 3 + c];
      dsum += (2.f * inter + 1e-6f) / (upv + ohv + 1e-6f);
    }
  float dice = 1.f - dsum / (float)(BATCH * NCLS);
  float focal = acc[2] / (acc[3] + 1e-6f);
  float nvalid = 0.f, pensum = 0.f, ttot = 0.f;
  for (int b = 0; b < BATCH; ++b) {
    float tb = acc[104 + b * 3 + 2];   // per-batch count of target==2 pixels
    ttot += tb;
    if (tb > 0.f) { nvalid += 1.f; pensum += acc[152 + b]; }
  }
  float pen = (nvalid > 0.f) ? (pensum / fmaxf(nvalid * 2.f, 1.f)) : 0.f;
  float sep = ((ttot > 0.f) && (acc[4] > 0.f)) ? pen : 0.f;   // SEP_PW = 1.0
  out[0] = ce + 0.5f * dice + 0.5f * focal + 0.3f * sep;
}

extern "C" void kernel_launch(void* const* d_in, const int* in_sizes, int n_in,
                              void* d_out, int out_size, void* d_ws, size_t ws_size,
                              hipStream_t stream) {
  const float* pred = (const float*)d_in[0];
  const int*   tgt  = (const int*)d_in[1];
  const float* cw   = (const float*)d_in[2];
  float* out = (float*)d_out;

  char* ws = (char*)d_ws;
  float* acc  = (float*)ws;                        // 4 KB accumulator region
  int* t_lab  = (int*)(ws + 4096);
  int* p_lab  = t_lab + (size_t)BATCH * HW;        // contiguous: [t_lab | p_lab]
  int* vmx_t  = p_lab + (size_t)BATCH * HW;
  int* vmn_t  = vmx_t + (size_t)BATCH * HW;
  int* vmx_p  = vmn_t + (size_t)BATCH * HW;
  int* vmn_p  = vmx_p + (size_t)BATCH * HW;

  k_zero<<<1, 256, 0, stream>>>(acc);
  k_init_pairs<<<dim3(HW / 256, BATCH), 256, 0, stream>>>(vmx_t, vmn_t, vmx_p, vmn_p);
  k_main<<<dim3(32, BATCH), 256, 0, stream>>>(pred, tgt, cw, acc, t_lab, p_lab);
  for (int i = 0; i < GLOBAL_ITERS; ++i)
    k_cc<<<dim3(256, 2 * BATCH), 256, 0, stream>>>(t_lab);   // t & p masks
  k_scatter<<<dim3(HW / 256, BATCH), 256, 0, stream>>>(t_lab, p_lab,
                                                       vmx_t, vmn_t, vmx_p, vmn_p);
  k_count<<<dim3(HW / 256, BATCH, 2), 256, 0, stream>>>(vmx_t, vmn_t, vmx_p, vmn_p, acc);
  k_final<<<1, 64, 0, stream>>>(acc, out);
}